// GraphConvolution_41343355191813
// MI455X (gfx1250) — compile-verified
//
#include <hip/hip_runtime.h>
#include <hip/hip_bf16.h>

typedef __attribute__((ext_vector_type(2))) float v2f;
typedef __attribute__((ext_vector_type(8))) float v8f;

#define N_NODES 100000
#define N_EDGES 3200000
#define D       256
#define EPB     256   // edges per block in scatter phase

// ---------------------------------------------------------------------------
// Phase 1: agg[r, :] += v * x[c, :]  (edge-parallel scatter-add)
// Block = 256 threads = 4 edge slots x 64 threads; each thread covers 4 dims
// via a float4 gather (global_load_b128) and 4 hardware f32 atomics.
// x (102MB) and agg (102MB) are both L2-resident on MI455X (192MB L2), so
// this phase runs at L2/atomic throughput, not HBM.
// ---------------------------------------------------------------------------
__global__ __launch_bounds__(256) void gcn_scatter(
    const float* __restrict__ x,
    const int*   __restrict__ erow,
    const int*   __restrict__ ecol,
    const float* __restrict__ eval,
    float*       __restrict__ agg)
{
    const int sub = threadIdx.x >> 6;          // 0..3  edge sub-slot
    const int d4  = (threadIdx.x & 63) << 2;   // dim base 0..252 step 4

    int e = blockIdx.x * EPB + sub;
    #pragma unroll 4
    for (int i = 0; i < EPB / 4; ++i, e += 4) {
        const int   r = erow[e];
        const int   c = ecol[e];
        const float v = eval[e];
        const float4 xv = *(const float4*)(x + c * D + d4);
        float* dst = agg + r * D + d4;
        unsafeAtomicAdd(dst + 0, v * xv.x);
        unsafeAtomicAdd(dst + 1, v * xv.y);
        unsafeAtomicAdd(dst + 2, v * xv.z);
        unsafeAtomicAdd(dst + 3, v * xv.w);
    }
}

// ---------------------------------------------------------------------------
// Phase 2: out = agg @ W^T + b   with V_WMMA_F32_16X16X4_F32 (fp32 matrix op).
// One wave32 computes one 16x16 tile of out; K=256 in steps of 4.
//
// A (16x4 f32) per ISA layout: lane l<16 holds (M=l, K={k,k+1}),
//                              lane l>=16 holds (M=l-16, K={k+2,k+3})
//   -> float2 load at agg[(tm + l%16)*D + k + 2*(l/16)]
// B (4x16) = W^T tile, transpose-symmetric layout:
//   lane l<16 holds (K={k,k+1}, N=l); l>=16 holds (K={k+2,k+3}, N=l-16)
//   -> float2 load at W[(tn + l%16)*D + k + 2*(l/16)]   (same addressing!)
// C/D (16x16 f32): VGPR j -> (M = j + 8*(l/16), N = l%16)
// ---------------------------------------------------------------------------
__global__ __launch_bounds__(256) void gcn_gemm_wmma(
    const float* __restrict__ agg,
    const float* __restrict__ W,
    const float* __restrict__ bias,
    float*       __restrict__ out)
{
    const int wave = threadIdx.x >> 5;                 // 0..7
    const int lane = threadIdx.x & 31;
    const int tile = blockIdx.x * 8 + wave;            // wave-uniform
    const int tiles_n = D / 16;                        // 16
    const int tm = (tile / tiles_n) * 16;              // node-row base
    const int tn = (tile % tiles_n) * 16;              // out-col base
    if (tm >= N_NODES) return;                         // uniform per wave

    const int lm    = lane & 15;
    const int khalf = (lane >> 4) << 1;                // 0 or 2

    const float* arow = agg + (tm + lm) * D + khalf;
    const float* brow = W   + (tn + lm) * D + khalf;

    v8f c = {};
    #pragma unroll 8
    for (int k = 0; k < D; k += 4) {
        v2f a = *(const v2f*)(arow + k);
        v2f b = *(const v2f*)(brow + k);
        // 8 args: (neg_a, A, neg_b, B, c_mod, C, reuse_a, reuse_b)
        c = __builtin_amdgcn_wmma_f32_16x16x4_f32(
                false, a, false, b, (short)0, c, false, false);
    }

    const int mrow = tm + ((lane >> 4) << 3);          // +0 or +8
    const int ocol = tn + lm;
    const float bb = bias[ocol];
    #pragma unroll
    for (int j = 0; j < 8; ++j) {
        out[(mrow + j) * D + ocol] = c[j] + bb;
    }
}

extern "C" void kernel_launch(void* const* d_in, const int* in_sizes, int n_in,
                              void* d_out, int out_size, void* d_ws, size_t ws_size,
                              hipStream_t stream) {
    const float* x    = (const float*)d_in[0];
    const int*   erow = (const int*)  d_in[1];
    const int*   ecol = (const int*)  d_in[2];
    const float* eval = (const float*)d_in[3];
    const float* W    = (const float*)d_in[4];
    const float* b    = (const float*)d_in[5];
    float*       out  = (float*)d_out;
    float*       agg  = (float*)d_ws;   // N_NODES * D fp32 = 102.4 MB scratch

    // zero the accumulator (d_ws is poisoned by the harness)
    hipMemsetAsync(agg, 0, (size_t)N_NODES * D * sizeof(float), stream);

    // Phase 1: edge scatter-add
    gcn_scatter<<<N_EDGES / EPB, 256, 0, stream>>>(x, erow, ecol, eval, agg);

    // Phase 2: fp32 WMMA GEMM + bias
    // tiles = (100000/16) * (256/16) = 100000; 8 waves/block -> 12500 blocks
    const int n_tiles = (N_NODES / 16) * (D / 16);
    gcn_gemm_wmma<<<n_tiles / 8, 256, 0, stream>>>(agg, W, b, out);
}